// SimplePlasticLSTM_14508399526004
// MI455X (gfx1250) — compile-verified
//
#include <hip/hip_runtime.h>
#include <hip/hip_bf16.h>

// ---------------------------------------------------------------------------
// SimplePlasticLSTM step for MI455X (gfx1250).
//
// Roofline: hebb tensor is [256,512,512] fp32 = 268 MB; it must be read for
// h2c, read for the update, and hebb_new written: ~800 MB => ~35us at
// 23.3 TB/s. All GEMMs together are ~1 GFLOP and negligible, so we keep them
// in exact fp32 using V_WMMA_F32_16X16X4_F32 (one 16x16 tile per wave32).
// The hebb passes are pure float4-coalesced streams.
// ---------------------------------------------------------------------------

#define BN 256   // batch
#define IN 512   // input dim
#define HN 512   // hidden dim
#define BH (BN * HN)

typedef __attribute__((ext_vector_type(2))) float v2f;
typedef __attribute__((ext_vector_type(8))) float v8f;

__device__ __forceinline__ v8f wmma_f32_4(v2f a, v2f b, v8f c) {
  // 8 args: (neg_a, A, neg_b, B, c_mod, C, reuse_a, reuse_b)
  return __builtin_amdgcn_wmma_f32_16x16x4_f32(
      false, a, false, b, (short)0, c, false, false);
}

__device__ __forceinline__ float sigmoidf_dev(float x) {
  return 1.0f / (1.0f + __expf(-x));
}

// ---------------------------------------------------------------------------
// Generic 16x16-tile WMMA fp32 GEMM:  out = act( X1@op(W1) + b1 [+ X2@W2^T + b2] )
//   TRANSW=false : W1 is [N,K] row-major, computes X1 @ W1^T (torch Linear)
//   TRANSW=true  : W1 is [K,N] row-major, computes X1 @ W1
// M=BN, N=HN, one wave per 16x16 tile, 4 waves per block.
// ---------------------------------------------------------------------------
template <bool TRANSW, bool HAS2, bool SIG, bool HASB>
__global__ void __launch_bounds__(128)
gemm16_wmma(const float* __restrict__ X1, const float* __restrict__ W1,
            const float* __restrict__ b1, int K1,
            const float* __restrict__ X2, const float* __restrict__ W2,
            const float* __restrict__ b2, int K2,
            float* __restrict__ out, int M, int N) {
  const int wave = threadIdx.x >> 5;
  const int lane = threadIdx.x & 31;
  const int tile = blockIdx.x * 4 + wave;
  const int ntn  = N >> 4;
  const int m0   = (tile / ntn) << 4;
  const int n0   = (tile % ntn) << 4;
  const int half = lane >> 4;   // 0: K={0,1}, rows 0..7 of C ; 1: K={2,3}, rows 8..15
  const int l    = lane & 15;

  v8f acc = {};

  // ---- panel 1 ----
  {
    const float* Ap = X1 + (m0 + l) * K1 + 2 * half;
    const float* Bp = TRANSW ? W1 : (W1 + (n0 + l) * K1 + 2 * half);
#pragma unroll 8
    for (int k0 = 0; k0 < K1; k0 += 4) {
      v2f a = *(const v2f*)(Ap + k0);
      v2f b;
      if (!TRANSW) {
        b = *(const v2f*)(Bp + k0);
      } else {
        const int kk = k0 + 2 * half;
        b.x = Bp[(size_t)kk * N + n0 + l];
        b.y = Bp[(size_t)(kk + 1) * N + n0 + l];
      }
      acc = wmma_f32_4(a, b, acc);
    }
  }

  // ---- panel 2 (recurrent term for gates) ----
  if (HAS2) {
    const float* Ap = X2 + (m0 + l) * K2 + 2 * half;
    const float* Bp = W2 + (n0 + l) * K2 + 2 * half;
#pragma unroll 8
    for (int k0 = 0; k0 < K2; k0 += 4) {
      v2f a = *(const v2f*)(Ap + k0);
      v2f b = *(const v2f*)(Bp + k0);
      acc = wmma_f32_4(a, b, acc);
    }
  }

  float bs = 0.0f;
  if (HASB) {
    bs = b1[n0 + l];
    if (HAS2) bs += b2[n0 + l];
  }
#pragma unroll
  for (int r = 0; r < 8; ++r) {
    float v = acc[r] + bs;
    if (SIG) v = sigmoidf_dev(v);
    out[(size_t)(m0 + half * 8 + r) * N + n0 + l] = v;
  }
}

// ---------------------------------------------------------------------------
// Pass 1 over hebb: hpart[s][b][k] = sum_{h in split s} h0[b,h] * hebb[b,h,k]
// grid (BN, 4), 128 threads, 4 k-columns (float4) per thread. Coalesced b128.
// ---------------------------------------------------------------------------
__global__ void __launch_bounds__(128)
hebbmv_kernel(const float* __restrict__ h0, const float* __restrict__ hebb,
              float* __restrict__ hpart) {
  const int b = blockIdx.x;
  const int s = blockIdx.y;          // h split: [s*128, s*128+128)
  const int t = threadIdx.x;         // k4 index: k = 4t
  const float* h0b = h0 + (size_t)b * HN + s * 128;
  const float4* base =
      (const float4*)(hebb + ((size_t)b * HN + (size_t)s * 128) * HN) + t;

  float4 acc = {0.f, 0.f, 0.f, 0.f};
#pragma unroll 4
  for (int hh = 0; hh < 128; ++hh) {
    __builtin_prefetch(&base[(hh + 6) * (HN / 4)], 0, 1);  // global_prefetch_b8
    const float a = h0b[hh];                                // uniform -> SMEM
    const float4 v = base[hh * (HN / 4)];
    acc.x += a * v.x;
    acc.y += a * v.y;
    acc.z += a * v.z;
    acc.w += a * v.w;
  }
  ((float4*)(hpart + ((size_t)s * BN + b) * HN))[t] = acc;
}

// ---------------------------------------------------------------------------
// Elementwise combine over [B,H]:
//   h2c  = hw + alpha[k] * sum_s hpart[s]
//   itc  = tanh(cpre + h2c)
//   cell = fgt*c0 + ipt*itc          -> d_out[BH..2BH)
//   hact = opt*tanh(cell)            -> d_out[0..BH)
// ---------------------------------------------------------------------------
__global__ void __launch_bounds__(256)
combine_kernel(const float* __restrict__ fgt, const float* __restrict__ ipt,
               const float* __restrict__ opt, const float* __restrict__ cpre,
               const float* __restrict__ hw, const float* __restrict__ hpart,
               const float* __restrict__ alpha, const float* __restrict__ c0,
               float* __restrict__ out_hactiv, float* __restrict__ out_cell,
               float* __restrict__ itc_ws) {
  const int idx = blockIdx.x * 256 + threadIdx.x;   // b*H + k
  const int k = idx & (HN - 1);
  const float hs = hpart[idx] + hpart[BH + idx] + hpart[2 * BH + idx] +
                   hpart[3 * BH + idx];
  const float h2c = hw[idx] + alpha[k] * hs;
  const float itc = tanhf(cpre[idx] + h2c);
  const float cell = fgt[idx] * c0[idx] + ipt[idx] * itc;
  const float ha = opt[idx] * tanhf(cell);
  out_hactiv[idx] = ha;
  out_cell[idx] = cell;
  itc_ws[idx] = itc;
}

// ---------------------------------------------------------------------------
// myeta[b] = tanh(hactiv[b,:] . Wmod + bmod);  mi[b,k] = (myeta*Wfan[k]+bfan[k])*itc[b,k]
// one block per sample, 256-thread tree reduction in LDS.
// ---------------------------------------------------------------------------
__global__ void __launch_bounds__(256)
myeta_kernel(const float* __restrict__ hactiv, const float* __restrict__ Wmod,
             const float* __restrict__ bmod, const float* __restrict__ Wfan,
             const float* __restrict__ bfan, const float* __restrict__ itc,
             float* __restrict__ mi) {
  __shared__ float red[256];
  const int b = blockIdx.x;
  const int t = threadIdx.x;
  const float* hb = hactiv + (size_t)b * HN;
  red[t] = hb[t] * Wmod[t] + hb[t + 256] * Wmod[t + 256];
  __syncthreads();
#pragma unroll
  for (int s = 128; s > 0; s >>= 1) {
    if (t < s) red[t] += red[t + s];
    __syncthreads();
  }
  const float eta = tanhf(red[0] + bmod[0]);
#pragma unroll
  for (int k = t; k < HN; k += 256) {
    mi[(size_t)b * HN + k] = (eta * Wfan[k] + bfan[k]) * itc[(size_t)b * HN + k];
  }
}

// ---------------------------------------------------------------------------
// Pass 2 over hebb: hebb_new = clip(hebb + h0[b,h] * mi[b,k], -2, 2)
// pure float4 stream: 16.7M float4 threads.
// ---------------------------------------------------------------------------
__global__ void __launch_bounds__(256)
hebbupd_kernel(const float* __restrict__ hebb, const float* __restrict__ h0,
               const float* __restrict__ mi, float* __restrict__ out_hebb) {
  const size_t i4 = (size_t)blockIdx.x * 256 + threadIdx.x;  // float4 index
  const size_t e = i4 * 4;
  const int b = (int)(e >> 18);          // / (H*H)
  const int rem = (int)(e & ((HN * HN) - 1));
  const int h = rem >> 9;                // / H
  const int k = rem & (HN - 1);
  const float a = h0[(size_t)b * HN + h];
  const float4 m = *(const float4*)(mi + (size_t)b * HN + k);
  const float4 v = ((const float4*)hebb)[i4];
  float4 r;
  r.x = fminf(2.0f, fmaxf(-2.0f, v.x + a * m.x));
  r.y = fminf(2.0f, fmaxf(-2.0f, v.y + a * m.y));
  r.z = fminf(2.0f, fmaxf(-2.0f, v.z + a * m.z));
  r.w = fminf(2.0f, fmaxf(-2.0f, v.w + a * m.w));
  ((float4*)out_hebb)[i4] = r;
}

// ---------------------------------------------------------------------------
extern "C" void kernel_launch(void* const* d_in, const int* in_sizes, int n_in,
                              void* d_out, int out_size, void* d_ws,
                              size_t ws_size, hipStream_t stream) {
  const float* inputs = (const float*)d_in[0];
  const float* h0     = (const float*)d_in[1];
  const float* c0     = (const float*)d_in[2];
  const float* hebb   = (const float*)d_in[3];
  const float* w      = (const float*)d_in[4];
  const float* alpha  = (const float*)d_in[5];
  const float* Wxf = (const float*)d_in[6],  *bxf = (const float*)d_in[7];
  const float* Whf = (const float*)d_in[8],  *bhf = (const float*)d_in[9];
  const float* Wxi = (const float*)d_in[10], *bxi = (const float*)d_in[11];
  const float* Whi = (const float*)d_in[12], *bhi = (const float*)d_in[13];
  const float* Wxo = (const float*)d_in[14], *bxo = (const float*)d_in[15];
  const float* Who = (const float*)d_in[16], *bho = (const float*)d_in[17];
  const float* Wxc = (const float*)d_in[18], *bxc = (const float*)d_in[19];
  const float* Wmod = (const float*)d_in[20], *bmod = (const float*)d_in[21];
  const float* Wfan = (const float*)d_in[22], *bfan = (const float*)d_in[23];

  float* out_hactiv = (float*)d_out;            // [B,H]
  float* out_cell   = (float*)d_out + BH;       // [B,H]
  float* out_hebb   = (float*)d_out + 2 * BH;   // [B,H,H]

  float* ws    = (float*)d_ws;
  float* fgt   = ws + 0 * BH;
  float* ipt   = ws + 1 * BH;
  float* opt   = ws + 2 * BH;
  float* cpre  = ws + 3 * BH;
  float* hw    = ws + 4 * BH;
  float* hpart = ws + 5 * BH;   // 4 * BH
  float* itc   = ws + 9 * BH;
  float* mi    = ws + 10 * BH;

  const int gemm_blocks = (BN / 16) * (HN / 16) / 4;  // 128 blocks x 4 waves

  // gates: sigmoid(inputs@Wx^T + bx + h0@Wh^T + bh)
  gemm16_wmma<false, true, true, true><<<gemm_blocks, 128, 0, stream>>>(
      inputs, Wxf, bxf, IN, h0, Whf, bhf, HN, fgt, BN, HN);
  gemm16_wmma<false, true, true, true><<<gemm_blocks, 128, 0, stream>>>(
      inputs, Wxi, bxi, IN, h0, Whi, bhi, HN, ipt, BN, HN);
  gemm16_wmma<false, true, true, true><<<gemm_blocks, 128, 0, stream>>>(
      inputs, Wxo, bxo, IN, h0, Who, bho, HN, opt, BN, HN);
  // cell input pre-activation (no recurrent weight term yet, no activation)
  gemm16_wmma<false, false, false, true><<<gemm_blocks, 128, 0, stream>>>(
      inputs, Wxc, bxc, IN, nullptr, nullptr, nullptr, 0, cpre, BN, HN);
  // hw = h0 @ w  (w is [H,H], not transposed)
  gemm16_wmma<true, false, false, false><<<gemm_blocks, 128, 0, stream>>>(
      h0, w, nullptr, HN, nullptr, nullptr, nullptr, 0, hw, BN, HN);

  // pass 1 over hebb (268 MB read): per-sample reduction, 4-way h split
  hebbmv_kernel<<<dim3(BN, 4), 128, 0, stream>>>(h0, hebb, hpart);

  // fuse: h2c, inputstocell, cell, hactiv
  combine_kernel<<<BH / 256, 256, 0, stream>>>(
      fgt, ipt, opt, cpre, hw, hpart, alpha, c0, out_hactiv, out_cell, itc);

  // neuromodulated learning rate, premultiplied by itc
  myeta_kernel<<<BN, 256, 0, stream>>>(out_hactiv, Wmod, bmod, Wfan, bfan,
                                       itc, mi);

  // pass 2 over hebb (268 MB read + 268 MB write)
  hebbupd_kernel<<<(BN * HN * HN / 4) / 256, 256, 0, stream>>>(hebb, h0, mi,
                                                               out_hebb);
}